// BondEnergyModule_14482629722859
// MI455X (gfx1250) — compile-verified
//
#include <hip/hip_runtime.h>

// Harmonic bond energy: out[i] = par[i] * (|xyz[a_i] - xyz[b_i]| - len[i])^2
//
// Memory-bound gather kernel for MI455X (gfx1250, wave32):
//  - streamed operands (adj/len/par/out) use non-temporal cache hints (th:NT)
//    so the 12 MB xyz gather table stays resident in the 192 MB L2
//  - 4 bonds per thread; all streamed traffic moves as b128 per instruction
//    in s_clause groups
//  - gathers use 32-bit byte offsets (idx*12 < 2^32) -> GVS addressing
//    (SGPR base + 32-bit VGPR offset), one v_mul per address instead of a
//    64-bit mad pair; xyz reads keep the default RT cache policy (L2 hits)
//  - raw v_sqrt_f32 (1 ulp) instead of the IEEE sqrt fix-up sequence

typedef float     v4f __attribute__((ext_vector_type(4)));
typedef long long v2l __attribute__((ext_vector_type(2)));

#define BPT 4               // bonds per thread
#define BLOCK 256           // 8 wave32 waves per workgroup

__global__ __launch_bounds__(BLOCK) void bond_energy_kernel(
    const float* __restrict__ xyz,        // [n_atoms, 3] f32
    const long long* __restrict__ adj,    // [n_bonds, 2] i64 (values < 2^20)
    const float* __restrict__ blen,       // [n_bonds] f32
    const float* __restrict__ bpar,       // [n_bonds] f32
    float* __restrict__ out,              // [n_bonds] f32
    int n)
{
    const int base = (blockIdx.x * BLOCK + threadIdx.x) * BPT;
    const char* xyzb = (const char*)xyz;

    if (base + BPT <= n) {
        // Streaming loads: non-temporal (th:NT) to avoid polluting L2.
        v4f len4 = __builtin_nontemporal_load((const v4f*)(blen + base));
        v4f par4 = __builtin_nontemporal_load((const v4f*)(bpar + base));

        v4f res;
#pragma unroll
        for (int k = 0; k < BPT; ++k) {
            // One bond's index pair = 16 bytes -> global_load_b128 th:NT
            v2l ab = __builtin_nontemporal_load((const v2l*)(adj + 2ll * (base + k)));
            // 32-bit byte offsets: idx < 1M so idx*12 < 12M fits easily.
            unsigned oa = (unsigned)ab.x * 12u;
            unsigned ob = (unsigned)ab.y * 12u;
            const float* __restrict__ p0 = (const float*)(xyzb + oa);
            const float* __restrict__ p1 = (const float*)(xyzb + ob);
            // Gathers: regular temporal policy, expect L2 hits (xyz = 12 MB)
            float dx = p0[0] - p1[0];
            float dy = p0[1] - p1[1];
            float dz = p0[2] - p1[2];
            float e  = __builtin_amdgcn_sqrtf(dx * dx + dy * dy + dz * dz);
            float d  = e - len4[k];
            res[k]   = par4[k] * d * d;
        }
        // Streaming store: non-temporal b128
        __builtin_nontemporal_store(res, (v4f*)(out + base));
    } else if (base < n) {
        // Tail (never taken for n % 4 == 0, kept for generality)
        for (int k = 0; k < BPT; ++k) {
            int i = base + k;
            if (i < n) {
                unsigned oa = (unsigned)adj[2ll * i] * 12u;
                unsigned ob = (unsigned)adj[2ll * i + 1] * 12u;
                const float* p0 = (const float*)(xyzb + oa);
                const float* p1 = (const float*)(xyzb + ob);
                float dx = p0[0] - p1[0];
                float dy = p0[1] - p1[1];
                float dz = p0[2] - p1[2];
                float e  = __builtin_amdgcn_sqrtf(dx * dx + dy * dy + dz * dz);
                float d  = e - blen[i];
                out[i]   = bpar[i] * d * d;
            }
        }
    }
}

extern "C" void kernel_launch(void* const* d_in, const int* in_sizes, int n_in,
                              void* d_out, int out_size, void* d_ws, size_t ws_size,
                              hipStream_t stream)
{
    // setup_inputs() order: xyz (f32), bond_adj (i64), bond_len (f32), bond_par (f32)
    const float*     xyz  = (const float*)d_in[0];
    const long long* adj  = (const long long*)d_in[1];
    const float*     blen = (const float*)d_in[2];
    const float*     bpar = (const float*)d_in[3];
    float*           out  = (float*)d_out;

    const int n = out_size;  // one energy per bond
    const int bondsPerBlock = BLOCK * BPT;
    const int grid = (n + bondsPerBlock - 1) / bondsPerBlock;

    bond_energy_kernel<<<grid, BLOCK, 0, stream>>>(xyz, adj, blen, bpar, out, n);
}